// AttentionBlock_18468359373284
// MI455X (gfx1250) — compile-verified
//
#include <hip/hip_runtime.h>
#include <math.h>

typedef float v2f __attribute__((ext_vector_type(2)));
typedef float v8f __attribute__((ext_vector_type(8)));

#define CDIM 64
#define NPOS 4096          // 64*64 spatial positions
#define ATT_SCALE 0.125f   // C^-0.5
#define GN_EPS 1e-5f
#define NEG_BIG -1.0e30f

__device__ __forceinline__ v8f wmma_f32(v2f a, v2f b, v8f c) {
  // D = A(16x4 f32) * B(4x16 f32) + C(16x16 f32)
  return __builtin_amdgcn_wmma_f32_16x16x4_f32(false, a, false, b, (short)0, c,
                                               false, false);
}

// ---------------------------------------------------------------------------
// Kernel 1: GroupNorm over dim-1 (H). Each (b,g) group is a contiguous
// 32768-float chunk of x. gamma/beta are indexed by h (row).
// grid = 32 blocks (b*8+g), 256 threads.
// ---------------------------------------------------------------------------
__global__ void groupnorm_kernel(const float* __restrict__ x,
                                 const float* __restrict__ gamma,
                                 const float* __restrict__ beta,
                                 float* __restrict__ xn) {
  __shared__ float s_sum[256];
  __shared__ float s_sq[256];
  __shared__ float s_mean, s_inv;
  const int blk = blockIdx.x;        // b*8 + g
  const int g = blk & 7;
  const size_t base = (size_t)blk * 32768u;
  float sum = 0.f, sq = 0.f;
  for (int i = threadIdx.x; i < 32768; i += 256) {
    float v = x[base + i];
    sum += v;
    sq += v * v;
  }
  s_sum[threadIdx.x] = sum;
  s_sq[threadIdx.x] = sq;
  __syncthreads();
  for (int s = 128; s > 0; s >>= 1) {
    if ((int)threadIdx.x < s) {
      s_sum[threadIdx.x] += s_sum[threadIdx.x + s];
      s_sq[threadIdx.x] += s_sq[threadIdx.x + s];
    }
    __syncthreads();
  }
  if (threadIdx.x == 0) {
    float mean = s_sum[0] * (1.f / 32768.f);
    float var = s_sq[0] * (1.f / 32768.f) - mean * mean;
    s_mean = mean;
    s_inv = rsqrtf(var + GN_EPS);
  }
  __syncthreads();
  const float mean = s_mean, inv = s_inv;
  for (int i = threadIdx.x; i < 32768; i += 256) {
    int h = (g << 3) + (i >> 12);    // i / (64*64) rows within group
    xn[base + i] = (x[base + i] - mean) * inv * gamma[h] + beta[h];
  }
}

// ---------------------------------------------------------------------------
// Kernel 2: Q/K/V projections. [16384 x 64] @ [64 x 64] (+bias), x3 matrices.
// One wave per 16x16 output tile; 16 chained v_wmma_f32_16x16x4_f32 over K=64.
// grid = 1536 blocks x 256 threads (8 waves): 1024 M-tiles * 12 (mat,ntile).
// ---------------------------------------------------------------------------
__global__ void qkv_kernel(const float* __restrict__ xn,
                           const float* __restrict__ Wq, const float* __restrict__ bq,
                           const float* __restrict__ Wk, const float* __restrict__ bk,
                           const float* __restrict__ Wv, const float* __restrict__ bv,
                           float* __restrict__ q, float* __restrict__ k,
                           float* __restrict__ v) {
  const int wave = (blockIdx.x * blockDim.x + threadIdx.x) >> 5;
  const int lane = threadIdx.x & 31;
  const int mtile = wave / 12;
  const int which = wave % 12;
  const int mat = which >> 2;
  const int ntile = which & 3;
  const float* W;
  const float* bias;
  float* out;
  if (mat == 0)      { W = Wq; bias = bq; out = q; }
  else if (mat == 1) { W = Wk; bias = bk; out = k; }
  else               { W = Wv; bias = bv; out = v; }
  const int m0 = mtile << 4;
  const int n0 = ntile << 4;
  const int lrow = lane & 15;
  const int hi = lane >> 4;
  const int koff = hi << 1;        // lanes 0-15 -> K=kk..kk+1, 16-31 -> kk+2..3
  const float* arow = xn + (size_t)(m0 + lrow) * CDIM;
  v8f acc = {};
#pragma unroll
  for (int kk = 0; kk < CDIM; kk += 4) {
    v2f a = *(const v2f*)(arow + kk + koff);
    v2f b = { W[(kk + koff) * CDIM + n0 + lrow],
              W[(kk + koff + 1) * CDIM + n0 + lrow] };
    acc = wmma_f32(a, b, acc);
  }
  const float bb = bias[n0 + lrow];
#pragma unroll
  for (int r = 0; r < 8; r++) {
    int row = m0 + r + (hi << 3);  // D layout: lanes 0-15 M=r, 16-31 M=8+r
    out[(size_t)row * CDIM + n0 + lrow] = acc[r] + bb;
  }
}

// ---------------------------------------------------------------------------
// Kernel 3: flash attention. Block = (b, 64-query tile), 512 threads = 16
// waves, wave(mi,ni) owns S-tile (mi,ni) and O-tile (mi,ni). Online softmax
// with running m/l/alpha in LDS; scores never touch global memory.
// grid = 256 blocks x 512 threads.
// ---------------------------------------------------------------------------
#define SP_STRIDE 66                 // 64 + 2: keeps float2 LDS loads 8B-aligned

__global__ void attn_kernel(const float* __restrict__ q,
                            const float* __restrict__ k,
                            const float* __restrict__ v,
                            float* __restrict__ o) {
  __shared__ float s_p[64 * SP_STRIDE];
  __shared__ float s_m[64];
  __shared__ float s_l[64];
  __shared__ float s_alpha[64];
  const int b = blockIdx.x >> 6;
  const int qt = blockIdx.x & 63;
  const int tid = threadIdx.x;
  const int wave = tid >> 5;
  const int lane = tid & 31;
  const int mi = wave >> 2;
  const int ni = wave & 3;
  const int lrow = lane & 15;
  const int hi = lane >> 4;
  const int koff = hi << 1;
  const size_t bbase = (size_t)b * (NPOS * CDIM);

  // Preload this wave's Q A-fragments (row = query, K = channel).
  v2f af[16];
  {
    const float* qrow = q + bbase + (size_t)(qt * 64 + mi * 16 + lrow) * CDIM;
#pragma unroll
    for (int s = 0; s < 16; s++) af[s] = *(const v2f*)(qrow + 4 * s + koff);
  }
  if (tid < 64) { s_m[tid] = NEG_BIG; s_l[tid] = 0.f; }
  v8f acc = {};
  __syncthreads();

  for (int t = 0; t < 64; t++) {
    // --- S tile = Q(mi) @ K_t(ni)^T * scale ---------------------------------
    const float* krow = k + bbase + (size_t)(t * 64 + ni * 16 + lrow) * CDIM;
    v8f sacc = {};
#pragma unroll
    for (int s = 0; s < 16; s++) {
      v2f bf = *(const v2f*)(krow + 4 * s + koff);  // B col n = key row n
      sacc = wmma_f32(af[s], bf, sacc);
    }
#pragma unroll
    for (int r = 0; r < 8; r++) {
      int row = mi * 16 + r + (hi << 3);
      s_p[row * SP_STRIDE + ni * 16 + lrow] = sacc[r] * ATT_SCALE;
    }
    __syncthreads();

    // --- online softmax: 8 threads per row, shfl-width-8 reductions ---------
    {
      const int row = tid >> 3;
      const int sub = tid & 7;
      float* prow = s_p + row * SP_STRIDE + sub * 8;
      float lmax = NEG_BIG;
#pragma unroll
      for (int j = 0; j < 8; j++) lmax = fmaxf(lmax, prow[j]);
      for (int off = 1; off < 8; off <<= 1)
        lmax = fmaxf(lmax, __shfl_xor(lmax, off, 8));
      const float m_old = s_m[row];
      const float m_new = fmaxf(m_old, lmax);
      float lsum = 0.f;
#pragma unroll
      for (int j = 0; j < 8; j++) {
        float p = expf(prow[j] - m_new);
        prow[j] = p;                       // S becomes P in place
        lsum += p;
      }
      for (int off = 1; off < 8; off <<= 1) lsum += __shfl_xor(lsum, off, 8);
      if (sub == 0) {
        float alpha = expf(m_old - m_new);
        s_alpha[row] = alpha;
        s_m[row] = m_new;
        s_l[row] = s_l[row] * alpha + lsum;
      }
    }
    __syncthreads();

    // --- O = alpha * O + P @ V_t -------------------------------------------
#pragma unroll
    for (int r = 0; r < 8; r++) acc[r] *= s_alpha[mi * 16 + r + (hi << 3)];
    const float* vcol = v + bbase + (size_t)(t * 64) * CDIM + ni * 16 + lrow;
    const float* prow = s_p + (size_t)(mi * 16 + lrow) * SP_STRIDE;
#pragma unroll
    for (int s = 0; s < 16; s++) {
      v2f a = *(const v2f*)(prow + 4 * s + koff);   // P A-frag from LDS
      v2f bf = { vcol[(4 * s + koff) * CDIM],
                 vcol[(4 * s + koff + 1) * CDIM] }; // V B-frag, contraction=key
      acc = wmma_f32(a, bf, acc);
    }
    __syncthreads();  // protect s_p before next tile overwrites it
  }

  // Epilogue: normalize by running sum and store O.
#pragma unroll
  for (int r = 0; r < 8; r++) {
    int row = mi * 16 + r + (hi << 3);
    o[bbase + (size_t)(qt * 64 + row) * CDIM + ni * 16 + lrow] =
        acc[r] / s_l[row];
  }
}

// ---------------------------------------------------------------------------
// Kernel 4: out = xn + O @ Wo + bo.  grid = 512 blocks x 256 threads.
// ---------------------------------------------------------------------------
__global__ void outproj_kernel(const float* __restrict__ o,
                               const float* __restrict__ Wo,
                               const float* __restrict__ bo,
                               const float* __restrict__ xn,
                               float* __restrict__ out) {
  const int wave = (blockIdx.x * blockDim.x + threadIdx.x) >> 5;
  const int lane = threadIdx.x & 31;
  const int mtile = wave >> 2;
  const int ntile = wave & 3;
  const int m0 = mtile << 4;
  const int n0 = ntile << 4;
  const int lrow = lane & 15;
  const int hi = lane >> 4;
  const int koff = hi << 1;
  const float* arow = o + (size_t)(m0 + lrow) * CDIM;
  v8f acc = {};
#pragma unroll
  for (int kk = 0; kk < CDIM; kk += 4) {
    v2f a = *(const v2f*)(arow + kk + koff);
    v2f b = { Wo[(kk + koff) * CDIM + n0 + lrow],
              Wo[(kk + koff + 1) * CDIM + n0 + lrow] };
    acc = wmma_f32(a, b, acc);
  }
  const float bb = bo[n0 + lrow];
#pragma unroll
  for (int r = 0; r < 8; r++) {
    size_t idx = (size_t)(m0 + r + (hi << 3)) * CDIM + n0 + lrow;
    out[idx] = xn[idx] + acc[r] + bb;
  }
}

// ---------------------------------------------------------------------------
extern "C" void kernel_launch(void* const* d_in, const int* in_sizes, int n_in,
                              void* d_out, int out_size, void* d_ws,
                              size_t ws_size, hipStream_t stream) {
  const float* x     = (const float*)d_in[0];
  const float* gamma = (const float*)d_in[1];
  const float* beta  = (const float*)d_in[2];
  const float* Wq    = (const float*)d_in[3];
  const float* bq    = (const float*)d_in[4];
  const float* Wk    = (const float*)d_in[5];
  const float* bk    = (const float*)d_in[6];
  const float* Wv    = (const float*)d_in[7];
  const float* bv    = (const float*)d_in[8];
  const float* Wo    = (const float*)d_in[9];
  const float* bo    = (const float*)d_in[10];
  float* out = (float*)d_out;

  float* ws = (float*)d_ws;
  const size_t SZ = (size_t)4 * NPOS * CDIM;  // 1,048,576 floats per buffer
  float* xn = ws;
  float* q  = ws + SZ;
  float* k  = ws + 2 * SZ;
  float* v  = ws + 3 * SZ;
  float* o  = ws + 4 * SZ;

  groupnorm_kernel<<<32, 256, 0, stream>>>(x, gamma, beta, xn);
  qkv_kernel<<<1536, 256, 0, stream>>>(xn, Wq, bq, Wk, bk, Wv, bv, q, k, v);
  attn_kernel<<<256, 512, 0, stream>>>(q, k, v, o);
  outproj_kernel<<<512, 256, 0, stream>>>(o, Wo, bo, xn, out);
}